// Ive_38835094290611
// MI455X (gfx1250) — compile-verified
//
#include <hip/hip_runtime.h>
#include <cstdint>
#include <cmath>

static constexpr int    NSTEPS = 256;   // recurrence steps -> terms 1..256 (+term 0)
static constexpr int    NGROUP = 32;    // 32 groups x 8 steps
static constexpr double VORD   = 63.0;  // Bessel order v = 128/2 - 1

// ---------------------------------------------------------------------------
// Init: grouped prefix-product table + lgamma(64) in d_ws (f64).
//   For step k (term k+1), group g = k/8:
//     P[k] = prod_{m=8g..k} 1/((m+1)(m+64))
//   ws[256] = lgamma(64) = log(63!) by exact f64 log-summation.
// ---------------------------------------------------------------------------
__global__ __launch_bounds__(256) void ive_init_tables(double* __restrict__ ws) {
    const int k    = threadIdx.x;
    const int base = k & ~7;
    double p = 1.0;
    for (int m = base; m <= k; ++m)
        p *= 1.0 / ((double)(m + 1) * (double)(m + 64));
    ws[k] = p;
    if (k == 0) {
        double s = 0.0;
        for (int i = 2; i <= 63; ++i) s += log((double)i);
        ws[NSTEPS] = s;
    }
}

// ---------------------------------------------------------------------------
// Core: ive(63,z) = e^{-z} I_63(z), scaled-term recurrence in f64.
//   t0 = exp(63*log(z/2) - lgamma(64) - z)   (e^{-z} folded in)
//   term_{8g+i+1} = t_{8g} * u^{i+1} * P[8g+i],  u = (z/2)^2
// 8-term batching: 17 f64 ops / 8 terms / element; t-chain advances once per
// group. 4 elements x 2 accumulators = 8 independent fma chains.
// Wave-uniform LDS reads broadcast; early exit once all lanes' tails < 1e-18.
// ---------------------------------------------------------------------------
__device__ __attribute__((always_inline)) void ive_compute_body(
        const float* __restrict__ z, float* __restrict__ out,
        const double* s_tab, double lg64, int n) {
    const int tid = threadIdx.x;
    const int i0  = blockIdx.x * (256 * 4) + tid;

    double t[4], accA[4], accB[4], up[4][8];
#pragma unroll
    for (int j = 0; j < 4; ++j) {
        const int    idx = i0 + j * 256;
        const float  zf  = (idx < n) ? z[idx] : 1.0f;
        const double zd  = (double)zf;
        const double lhz = log(0.5 * zd);
        t[j]    = exp(VORD * lhz - lg64 - zd);  // scaled term_0
        accA[j] = t[j];
        accB[j] = 0.0;
        const double uu = 0.25 * zd * zd;       // (z/2)^2
        up[j][0] = uu;
#pragma unroll
        for (int i = 1; i < 8; ++i) up[j][i] = up[j][i - 1] * uu;
    }

#pragma unroll 1
    for (int g = 0; g < NGROUP; ++g) {
        double w[8];
#pragma unroll
        for (int i = 0; i < 8; ++i) w[i] = s_tab[g * 8 + i];  // LDS broadcast

        int live = 0;
#pragma unroll
        for (int j = 0; j < 4; ++j) {
            const double tj = t[j];
            accA[j] = fma(tj, up[j][0] * w[0], accA[j]);
            accB[j] = fma(tj, up[j][1] * w[1], accB[j]);
            accA[j] = fma(tj, up[j][2] * w[2], accA[j]);
            accB[j] = fma(tj, up[j][3] * w[3], accB[j]);
            accA[j] = fma(tj, up[j][4] * w[4], accA[j]);
            accB[j] = fma(tj, up[j][5] * w[5], accB[j]);
            accA[j] = fma(tj, up[j][6] * w[6], accA[j]);
            const double pl = up[j][7] * w[7];
            accB[j] = fma(tj, pl, accB[j]);
            t[j]    = tj * pl;                       // advance 8 terms at once
            live   |= (t[j] > accA[j] * 1e-18) ? 1 : 0;
        }
        if (!__any(live)) break;   // whole wave converged (tail < 1e-18 rel)
    }

#pragma unroll
    for (int j = 0; j < 4; ++j) {
        const int idx = i0 + j * 256;
        if (idx < n) out[idx] = (float)(accA[j] + accB[j]);
    }
}

// ---------------------------------------------------------------------------
// Main kernel: stage the 257-double table into LDS via the CDNA5 async
// global->LDS data mover (ASYNCcnt), then run the series.
// ---------------------------------------------------------------------------
__global__ __launch_bounds__(256) void ive_kernel_async(
        const float* __restrict__ z, float* __restrict__ out,
        const double* __restrict__ tbl, int n) {
    __shared__ double s_tab[NSTEPS + 1];
    const int tid = threadIdx.x;

    {
        // Low 32 bits of a generic LDS pointer = LDS byte offset (ISA 10.2).
        const uint32_t lbase = (uint32_t)(uintptr_t)(&s_tab[0]);
        const uint64_t gbase = (uint64_t)(uintptr_t)tbl;
        const uint32_t loff  = lbase + (uint32_t)tid * 8u;
        const uint64_t goff  = gbase + (uint64_t)tid * 8u;
        asm volatile("global_load_async_to_lds_b64 %0, %1, off"
                     :: "v"(loff), "v"(goff) : "memory");
        if (tid == 0) {   // extra slot: lgamma(64)
            const uint32_t l2 = lbase + (uint32_t)NSTEPS * 8u;
            const uint64_t g2 = gbase + (uint64_t)NSTEPS * 8u;
            asm volatile("global_load_async_to_lds_b64 %0, %1, off"
                         :: "v"(l2), "v"(g2) : "memory");
        }
        asm volatile("s_wait_asynccnt 0" ::: "memory");
    }
    __syncthreads();

    ive_compute_body(z, out, s_tab, s_tab[NSTEPS], n);
}

// Fallback if d_ws is too small: build the table per-block in LDS directly.
__global__ __launch_bounds__(256) void ive_kernel_selfinit(
        const float* __restrict__ z, float* __restrict__ out, int n) {
    __shared__ double s_tab[NSTEPS + 1];
    const int tid  = threadIdx.x;
    const int base = tid & ~7;
    double p = 1.0;
    for (int m = base; m <= tid; ++m)
        p *= 1.0 / ((double)(m + 1) * (double)(m + 64));
    s_tab[tid] = p;
    if (tid == 0) {
        double s = 0.0;
        for (int i = 2; i <= 63; ++i) s += log((double)i);
        s_tab[NSTEPS] = s;
    }
    __syncthreads();
    ive_compute_body(z, out, s_tab, s_tab[NSTEPS], n);
}

extern "C" void kernel_launch(void* const* d_in, const int* in_sizes, int n_in,
                              void* d_out, int out_size, void* d_ws, size_t ws_size,
                              hipStream_t stream) {
    const float* z   = (const float*)d_in[0];
    float*       out = (float*)d_out;
    const int    n   = in_sizes[0];

    const int per_block = 256 * 4;
    const int blocks    = (n + per_block - 1) / per_block;

    if (ws_size >= (size_t)(NSTEPS + 1) * sizeof(double)) {
        double* tbl = (double*)d_ws;
        ive_init_tables<<<1, 256, 0, stream>>>(tbl);
        ive_kernel_async<<<blocks, 256, 0, stream>>>(z, out, tbl, n);
    } else {
        ive_kernel_selfinit<<<blocks, 256, 0, stream>>>(z, out, n);
    }
}